// ClusterNet_62878321213502
// MI455X (gfx1250) — compile-verified
//
#include <hip/hip_runtime.h>
#include <math.h>

typedef __attribute__((ext_vector_type(2))) float v2f;
typedef __attribute__((ext_vector_type(8))) float v8f;

static __device__ __forceinline__ v8f wmma_f32_k4(v2f a, v2f b, v8f c) {
  // D = A(16x4 f32) x B(4x16 f32) + C(16x16 f32)
  return __builtin_amdgcn_wmma_f32_16x16x4_f32(false, a, false, b, (short)0, c,
                                               false, false);
}

// ---------------- conv1 + maxpool2 : x(1024,3,32,32) -> p1(1024,6,14,14) ----
__global__ void conv1_pool(const float* __restrict__ x, const float* __restrict__ w,
                           const float* __restrict__ bias, float* __restrict__ p1) {
  int idx = blockIdx.x * blockDim.x + threadIdx.x;
  const int TOT = 1024 * 6 * 14 * 14;
  if (idx >= TOT) return;
  int pw = idx % 14; int t = idx / 14;
  int ph = t % 14;   t /= 14;
  int oc = t % 6;    int b = t / 6;
  const float* xb = x + (size_t)b * 3 * 32 * 32;
  const float* wf = w + oc * 75;
  float mx = -INFINITY;
  for (int dy = 0; dy < 2; ++dy)
    for (int dx = 0; dx < 2; ++dx) {
      int oh = ph * 2 + dy, ow = pw * 2 + dx;
      float s = bias[oc];
      for (int ic = 0; ic < 3; ++ic)
        #pragma unroll
        for (int ky = 0; ky < 5; ++ky)
          #pragma unroll
          for (int kx = 0; kx < 5; ++kx)
            s += xb[ic * 1024 + (oh + ky) * 32 + (ow + kx)] *
                 wf[ic * 25 + ky * 5 + kx];
      mx = fmaxf(mx, s);
    }
  p1[idx] = mx;
}

// ------------- conv2 + relu + maxpool2 : wv1(1024,6,14,14)->p2(1024,16,5,5) -
__global__ void conv2_relu_pool(const float* __restrict__ wv1, const float* __restrict__ w,
                                const float* __restrict__ bias, float* __restrict__ p2) {
  int idx = blockIdx.x * blockDim.x + threadIdx.x;
  const int TOT = 1024 * 16 * 5 * 5;
  if (idx >= TOT) return;
  int pw = idx % 5; int t = idx / 5;
  int ph = t % 5;   t /= 5;
  int oc = t % 16;  int b = t / 16;
  const float* ib = wv1 + (size_t)b * 6 * 196;
  const float* wf = w + oc * 150;
  float mx = -INFINITY;
  for (int dy = 0; dy < 2; ++dy)
    for (int dx = 0; dx < 2; ++dx) {
      int oh = ph * 2 + dy, ow = pw * 2 + dx;   // 0..9
      float s = bias[oc];
      for (int ic = 0; ic < 6; ++ic)
        #pragma unroll
        for (int ky = 0; ky < 5; ++ky)
          #pragma unroll
          for (int kx = 0; kx < 5; ++kx)
            s += ib[ic * 196 + (oh + ky) * 14 + (ow + kx)] *
                 wf[ic * 25 + ky * 5 + kx];
      mx = fmaxf(mx, s);
    }
  p2[idx] = fmaxf(mx, 0.0f);   // relu commutes with max
}

// ---------------- per-sample squared norms over channel dim -----------------
__global__ void sumsq_chan(const float* __restrict__ p, int Bn, int C, int HW,
                           float* __restrict__ pn) {
  int m = blockIdx.x * blockDim.x + threadIdx.x;
  if (m >= Bn * HW) return;
  int b = m / HW, hw = m % HW;
  const float* pp = p + (size_t)b * C * HW + hw;
  float s = 0.f;
  for (int c = 0; c < C; ++c) { float v = pp[c * HW]; s += v * v; }
  pn[m] = s;
}

// ---------------- per-key squared norms (keys stored (NC, C) row-major) -----
__global__ void sumsq_rows(const float* __restrict__ k, int NC, int C,
                           float* __restrict__ kn) {
  int l = blockIdx.x * blockDim.x + threadIdx.x;
  if (l >= NC) return;
  const float* kr = k + (size_t)l * C;
  float s = 0.f;
  for (int c = 0; c < C; ++c) s += kr[c] * kr[c];
  kn[l] = s;
}

// ------ cluster logits via f32 WMMA: al = -sqrt(max(pn + kn - 2 p.k, 0)) ----
// Each wave computes a 16 x (16*NT) strip. WMMA computes only p.k; the two
// rank-1 norm terms are added at store time. Software-pipelined K-loop:
// fragments for step k+4 are fetched before the WMMAs of step k execute.
template <int NT>
__global__ void cluster_wmma(const float* __restrict__ p, const float* __restrict__ pn,
                             const float* __restrict__ kmat, const float* __restrict__ kn,
                             float* __restrict__ al, int Bn, int C, int HW, int NC) {
  int M = Bn * HW;                     // multiple of 16 for all stages
  int wave = blockIdx.x * (blockDim.x >> 5) + (threadIdx.x >> 5);
  int tilesN = NC / (16 * NT);         // NC multiple of 16*NT in all uses
  int tilesM = M >> 4;
  if (wave >= tilesM * tilesN) return; // wave-uniform
  int tm = wave / tilesN, tn = wave % tilesN;
  int row0 = tm << 4, col0 = (tn * NT) << 4;
  int lane = threadIdx.x & 31;
  int half = lane >> 4, lid = lane & 15;
  int r  = row0 + lid;
  int b  = r / HW, hw = r % HW;
  const float* pr = p + (size_t)b * C * HW + hw;
  const float* krow[NT];
  #pragma unroll
  for (int t = 0; t < NT; ++t)
    krow[t] = kmat + (size_t)(col0 + t * 16 + lid) * C;

  const v8f vz = {0.f, 0.f, 0.f, 0.f, 0.f, 0.f, 0.f, 0.f};
  v8f acc[NT];
  #pragma unroll
  for (int t = 0; t < NT; ++t) acc[t] = vz;

  int Cm = C & ~3;
  if (Cm > 0) {
    // prologue: fragments for k = 0
    int ka = half * 2;
    v2f av;
    av.x = pr[(size_t)ka * HW];
    av.y = pr[(size_t)(ka + 1) * HW];
    v2f bv[NT];
    #pragma unroll
    for (int t = 0; t < NT; ++t) {
      bv[t].x = krow[t][ka];
      bv[t].y = krow[t][ka + 1];
    }
    for (int k = 4; k < Cm; k += 4) {
      int kan = k + half * 2;
      v2f avn;                                  // fetch next step first
      avn.x = pr[(size_t)kan * HW];
      avn.y = pr[(size_t)(kan + 1) * HW];
      v2f bvn[NT];
      #pragma unroll
      for (int t = 0; t < NT; ++t) {
        bvn[t].x = krow[t][kan];
        bvn[t].y = krow[t][kan + 1];
      }
      #pragma unroll
      for (int t = 0; t < NT; ++t)              // consume current step
        acc[t] = wmma_f32_k4(av, bv[t], acc[t]);
      av = avn;
      #pragma unroll
      for (int t = 0; t < NT; ++t) bv[t] = bvn[t];
    }
    #pragma unroll
    for (int t = 0; t < NT; ++t)                // epilogue
      acc[t] = wmma_f32_k4(av, bv[t], acc[t]);
  }
  if (C & 3) {                                  // wave-uniform guarded tail
    int ka = Cm + half * 2;
    v2f av;
    av.x = (ka     < C) ? pr[(size_t)ka * HW]       : 0.f;
    av.y = (ka + 1 < C) ? pr[(size_t)(ka + 1) * HW] : 0.f;
    #pragma unroll
    for (int t = 0; t < NT; ++t) {
      v2f bv;
      bv.x = (ka     < C) ? krow[t][ka]     : 0.f;
      bv.y = (ka + 1 < C) ? krow[t][ka + 1] : 0.f;
      acc[t] = wmma_f32_k4(av, bv, acc[t]);
    }
  }
  float knv[NT];
  #pragma unroll
  for (int t = 0; t < NT; ++t) knv[t] = kn[col0 + t * 16 + lid];
  #pragma unroll
  for (int v = 0; v < 8; ++v) {
    int rr = row0 + v + half * 8;               // always < M
    float pnv = pn[rr];
    size_t base = (size_t)(rr / HW) * NC * HW + (rr % HW);
    #pragma unroll
    for (int t = 0; t < NT; ++t) {
      int cc = col0 + t * 16 + lid;             // always < NC
      float d2 = pnv + knv[t] - 2.f * acc[t][v];
      al[base + (size_t)cc * HW] = -sqrtf(fmaxf(d2, 0.f));
    }
  }
}

// ---------------- generic f32 WMMA GEMM: C = A @ op(B) + bias ---------------
// Requires: M % 16 == 0, K % 4 == 0, K >= 4 (true for all uses). N ragged OK.
// Software-pipelined K-loop (register double buffering).
__global__ void wmma_gemm(const float* __restrict__ A, int lda,
                          const float* __restrict__ B, int ldb, int bTrans,
                          const float* __restrict__ bias,
                          float* __restrict__ Cc, int ldc,
                          int M, int N, int K) {
  int wave = blockIdx.x * (blockDim.x >> 5) + (threadIdx.x >> 5);
  int tilesN = (N + 15) >> 4;
  int tilesM = M >> 4;
  if (wave >= tilesM * tilesN) return;
  int tm = wave / tilesN, tn = wave % tilesN;
  int row0 = tm << 4, col0 = tn << 4;
  int lane = threadIdx.x & 31;
  int half = lane >> 4, lid = lane & 15;
  int r = row0 + lid;
  int col = col0 + lid;
  int colc = (col < N) ? col : (N - 1);       // clamp: dead cols accumulate
  const float* ap = A + (size_t)r * lda;      // garbage, discarded at store
  const float* bp;
  size_t bstep;
  if (bTrans) { bp = B + (size_t)colc * ldb; bstep = 1; }
  else        { bp = B + colc;               bstep = (size_t)ldb; }

  v8f acc = {0.f, 0.f, 0.f, 0.f, 0.f, 0.f, 0.f, 0.f};
  {
    int ka = half * 2;                        // prologue (k = 0)
    v2f av, bv;
    av.x = ap[ka];
    av.y = ap[ka + 1];
    bv.x = bp[(size_t)ka * bstep];
    bv.y = bp[(size_t)(ka + 1) * bstep];
    for (int k = 4; k < K; k += 4) {
      int kan = k + half * 2;
      v2f avn, bvn;                           // fetch next step first
      avn.x = ap[kan];
      avn.y = ap[kan + 1];
      bvn.x = bp[(size_t)kan * bstep];
      bvn.y = bp[(size_t)(kan + 1) * bstep];
      acc = wmma_f32_k4(av, bv, acc);         // consume current step
      av = avn; bv = bvn;
    }
    acc = wmma_f32_k4(av, bv, acc);           // epilogue
  }
  #pragma unroll
  for (int v = 0; v < 8; ++v) {
    int rr = row0 + v + half * 8;
    int cc = col0 + lid;
    if (cc < N) {
      float o = acc[v];
      if (bias) o += bias[cc];
      Cc[(size_t)rr * ldc + cc] = o;
    }
  }
}

// ---------------- per-key {max, sum, sumsq} over all samples ----------------
// sk[l, m] = al[b, l, hw]  with m = b*HW + hw ; one block per key l
__global__ void stage_stats(const float* __restrict__ al, int Bn, int NC, int HW,
                            float* __restrict__ st) {
  int l = blockIdx.x;
  int Ns = Bn * HW;
  __shared__ float smax[256], ssum[256], ssq[256];
  float mx = -INFINITY, s = 0.f, sq = 0.f;
  for (int m = threadIdx.x; m < Ns; m += blockDim.x) {
    int b = m / HW, hw = m % HW;
    float v = al[(size_t)b * NC * HW + (size_t)l * HW + hw];
    mx = fmaxf(mx, v); s += v; sq += v * v;
  }
  smax[threadIdx.x] = mx; ssum[threadIdx.x] = s; ssq[threadIdx.x] = sq;
  __syncthreads();
  for (int off = 128; off > 0; off >>= 1) {
    if (threadIdx.x < off) {
      smax[threadIdx.x] = fmaxf(smax[threadIdx.x], smax[threadIdx.x + off]);
      ssum[threadIdx.x] += ssum[threadIdx.x + off];
      ssq[threadIdx.x]  += ssq[threadIdx.x + off];
    }
    __syncthreads();
  }
  if (threadIdx.x == 0) {
    st[l] = smax[0]; st[NC + l] = ssum[0]; st[2 * NC + l] = ssq[0];
  }
}

// ------- fused softmax over NC + value combine (small C: 6 or 16) -----------
template <int C>
__global__ void softmax_wv(const float* __restrict__ al, const float* __restrict__ v,
                           float* __restrict__ wv, int Bn, int NC, int HW) {
  int m = blockIdx.x * blockDim.x + threadIdx.x;
  if (m >= Bn * HW) return;
  int b = m / HW, hw = m % HW;
  const float* a = al + (size_t)b * NC * HW + hw;
  float mx = -INFINITY;
  for (int l = 0; l < NC; ++l) mx = fmaxf(mx, a[(size_t)l * HW]);
  float acc[C];
  #pragma unroll
  for (int c = 0; c < C; ++c) acc[c] = 0.f;
  float se = 0.f;
  for (int l = 0; l < NC; ++l) {
    float e = expf(a[(size_t)l * HW] - mx);
    se += e;
    const float* vr = v + (size_t)l * C;
    #pragma unroll
    for (int c = 0; c < C; ++c) acc[c] += e * vr[c];
  }
  float inv = 1.f / se;
  #pragma unroll
  for (int c = 0; c < C; ++c)
    wv[(size_t)b * C * HW + (size_t)c * HW + hw] = acc[c] * inv;
}

// --------- row softmax (stage 3): one wave per row of length NC -------------
__global__ void softmax_rows(const float* __restrict__ al, float* __restrict__ s3,
                             int NC, int rows) {
  int wave = blockIdx.x * (blockDim.x >> 5) + (threadIdx.x >> 5);
  int lane = threadIdx.x & 31;
  if (wave >= rows) return;
  const float* a = al + (size_t)wave * NC;
  float mx = -INFINITY;
  for (int l = lane; l < NC; l += 32) mx = fmaxf(mx, a[l]);
  for (int off = 16; off; off >>= 1) mx = fmaxf(mx, __shfl_xor(mx, off, 32));
  float se = 0.f;
  for (int l = lane; l < NC; l += 32) {
    float e = expf(a[l] - mx);
    s3[(size_t)wave * NC + l] = e;
    se += e;
  }
  for (int off = 16; off; off >>= 1) se += __shfl_xor(se, off, 32);
  float inv = 1.f / se;
  for (int l = lane; l < NC; l += 32) s3[(size_t)wave * NC + l] *= inv;
}

// ---------------- finalize ng losses ----------------------------------------
__global__ void finalize_ng(const float* __restrict__ st1, const float* __restrict__ st2,
                            const float* __restrict__ st3, float* __restrict__ out) {
  float l1 = 0.f, l2 = 0.f;
  const float* sts[3] = {st1, st2, st3};
  const int ncs[3] = {64, 128, 512};
  const float nss[3] = {200704.f, 25600.f, 1024.f};
  for (int s = 0; s < 3; ++s) {
    const float* st = sts[s];
    int NC = ncs[s];
    float Ns = nss[s];
    float msum = 0.f, smu = 0.f, smu2 = 0.f, svar = 0.f, tsum = 0.f, tsq = 0.f;
    for (int l = 0; l < NC; ++l) {
      msum += st[l];
      float mu = st[NC + l] / Ns;
      smu += mu; smu2 += mu * mu;
      svar += st[2 * NC + l] / Ns - mu * mu;
      tsum += st[NC + l]; tsq += st[2 * NC + l];
    }
    l1 += -(msum / (float)NC);
    float mm = smu / (float)NC;
    float varMeans = smu2 / (float)NC - mm * mm;
    float meanVars = svar / (float)NC;
    float eve = varMeans - meanVars;
    float gm = tsum / ((float)NC * Ns);
    float gv = tsq / ((float)NC * Ns) - gm * gm;
    l2 += eve / gv;
  }
  out[10240] = l1;
  out[10241] = l2;
}

// ---------------------------------------------------------------------------
extern "C" void kernel_launch(void* const* d_in, const int* in_sizes, int n_in,
                              void* d_out, int out_size, void* d_ws, size_t ws_size,
                              hipStream_t stream) {
  const float* x       = (const float*)d_in[0];
  const float* conv1_w = (const float*)d_in[1];
  const float* conv1_b = (const float*)d_in[2];
  const float* conv2_w = (const float*)d_in[3];
  const float* conv2_b = (const float*)d_in[4];
  const float* fc1_w   = (const float*)d_in[5];
  const float* fc1_b   = (const float*)d_in[6];
  const float* fc2_w   = (const float*)d_in[7];
  const float* fc2_b   = (const float*)d_in[8];
  const float* k1s     = (const float*)d_in[9];
  const float* k2s     = (const float*)d_in[10];
  const float* k3s     = (const float*)d_in[11];
  const float* v1s     = (const float*)d_in[12];
  const float* v2s     = (const float*)d_in[13];
  const float* v3s     = (const float*)d_in[14];
  float* out = (float*)d_out;
  float* ws  = (float*)d_ws;

  // workspace layout (floats)
  const size_t O_BIG = 0;                       // 12,845,056 (al1 / al2 / al3+s3)
  const size_t O_P1  = 12845056;                // 1,204,224
  const size_t O_WV1 = O_P1 + 1204224;
  const size_t O_P2  = O_WV1 + 1204224;         // 409,600
  const size_t O_WV2 = O_P2 + 409600;
  const size_t O_P3  = O_WV2 + 409600;          // 122,880
  const size_t O_WV3 = O_P3 + 122880;
  const size_t O_PN  = O_WV3 + 122880;          // 200,704
  const size_t O_KN  = O_PN + 200704;           // 512
  const size_t O_ST1 = O_KN + 512;              // 192
  const size_t O_ST2 = O_ST1 + 192;             // 384
  const size_t O_ST3 = O_ST2 + 384;             // 1536
  float* al  = ws + O_BIG;
  float* s3  = ws + O_BIG + 524288;             // probs, after al3 within big
  float* p1  = ws + O_P1;
  float* wv1 = ws + O_WV1;
  float* p2  = ws + O_P2;
  float* wv2 = ws + O_WV2;
  float* p3  = ws + O_P3;
  float* wv3 = ws + O_WV3;
  float* pn  = ws + O_PN;
  float* kn  = ws + O_KN;
  float* st1 = ws + O_ST1;
  float* st2 = ws + O_ST2;
  float* st3 = ws + O_ST3;

  const int BLK = 256;
  const int WPB = BLK / 32;
  auto cdiv = [](int a, int b) { return (a + b - 1) / b; };

  // ---- stage 1 ----
  conv1_pool<<<cdiv(1024 * 6 * 196, BLK), BLK, 0, stream>>>(x, conv1_w, conv1_b, p1);
  sumsq_chan<<<cdiv(1024 * 196, BLK), BLK, 0, stream>>>(p1, 1024, 6, 196, pn);
  sumsq_rows<<<cdiv(64, BLK), BLK, 0, stream>>>(k1s, 64, 6, kn);
  {
    int waves = (200704 / 16) * (64 / 64);       // NT=4 -> 16x64 strip per wave
    cluster_wmma<4><<<cdiv(waves, WPB), BLK, 0, stream>>>(p1, pn, k1s, kn, al,
                                                          1024, 6, 196, 64);
  }
  stage_stats<<<64, BLK, 0, stream>>>(al, 1024, 64, 196, st1);
  softmax_wv<6><<<cdiv(1024 * 196, BLK), BLK, 0, stream>>>(al, v1s, wv1, 1024, 64, 196);

  // ---- stage 2 ----
  conv2_relu_pool<<<cdiv(1024 * 16 * 25, BLK), BLK, 0, stream>>>(wv1, conv2_w, conv2_b, p2);
  sumsq_chan<<<cdiv(1024 * 25, BLK), BLK, 0, stream>>>(p2, 1024, 16, 25, pn);
  sumsq_rows<<<cdiv(128, BLK), BLK, 0, stream>>>(k2s, 128, 16, kn);
  {
    int waves = (25600 / 16) * (128 / 64);
    cluster_wmma<4><<<cdiv(waves, WPB), BLK, 0, stream>>>(p2, pn, k2s, kn, al,
                                                          1024, 16, 25, 128);
  }
  stage_stats<<<128, BLK, 0, stream>>>(al, 1024, 128, 25, st2);
  softmax_wv<16><<<cdiv(1024 * 25, BLK), BLK, 0, stream>>>(al, v2s, wv2, 1024, 128, 25);

  // ---- stage 3 ----
  {
    int waves = (1024 / 16) * cdiv(120, 16);
    wmma_gemm<<<cdiv(waves, WPB), BLK, 0, stream>>>(wv2, 400, fc1_w, 400, 1,
                                                    fc1_b, p3, 120, 1024, 120, 400);
  }
  sumsq_chan<<<cdiv(1024, BLK), BLK, 0, stream>>>(p3, 1024, 120, 1, pn);
  sumsq_rows<<<cdiv(512, BLK), BLK, 0, stream>>>(k3s, 512, 120, kn);
  {
    int waves = (1024 / 16) * (512 / 64);
    cluster_wmma<4><<<cdiv(waves, WPB), BLK, 0, stream>>>(p3, pn, k3s, kn, al,
                                                          1024, 120, 1, 512);
  }
  stage_stats<<<512, BLK, 0, stream>>>(al, 1024, 512, 1, st3);
  softmax_rows<<<cdiv(1024, WPB), BLK, 0, stream>>>(al, s3, 512, 1024);
  {
    int waves = (1024 / 16) * cdiv(120, 16);
    wmma_gemm<<<cdiv(waves, WPB), BLK, 0, stream>>>(s3, 512, v3s, 120, 0,
                                                    (const float*)nullptr, wv3, 120,
                                                    1024, 120, 512);
  }
  {
    int waves = (1024 / 16) * cdiv(10, 16);
    wmma_gemm<<<cdiv(waves, WPB), BLK, 0, stream>>>(wv3, 120, fc2_w, 120, 1,
                                                    fc2_b, out, 10, 1024, 10, 120);
  }
  finalize_ng<<<1, 1, 0, stream>>>(st1, st2, st3, out);
  (void)in_sizes; (void)n_in; (void)out_size; (void)ws_size;
}